// PolyHarmInterpolator_43404939493678
// MI455X (gfx1250) — compile-verified
//
#include <hip/hip_runtime.h>
#include <hip/hip_bf16.h>

// PolyHarmonic interpolation on gfx1250 via V_WMMA_F32_16X16X4_F32.
//
// Shapes (from reference setup_inputs): B=4, m=8192, n=4096, d=3, k=16.
// out[b,m,k] = phi(d2) @ w + [x,1] @ v,  phi(t) = max(t,eps)^1.5,
// d2 = ||x||^2 + ||c||^2 - 2 x.c^T.
//
// Mapping: one wave32 owns a 16-row m-tile and all k=16 output columns.
//   - A (16x4 f32, 2 VGPRs): rows (x0,x1,x2,1)    [x_pad, reused for both GEMMs]
//   - linear term:  acc  = WMMA(A_x, B_v)          B_v = v[b] (4x16 exactly)
//   - per n-chunk:  d2   = WMMA(A_x, B_c, C=xsq)   B_c = (-2c0,-2c1,-2c2,||c||^2)
//                   (C preloaded with ||x_row||^2 -> full d2 in ONE instruction)
//                   phi  = t*v_sqrt(t), t = v_med3(d2, eps, +MAX)  [1 op clamp]
//                   phi tile -> wave-private LDS (C-layout -> A-layout remap)
//                   acc  = WMMA(A_phi_kb, B_w_kb, acc)  for kb = 0,4,8,12
// 5 WMMAs per 16-wide n-chunk, 256 chunks/wave, 2048 waves total.
// All data (~3.6 MB) is L2-resident; kernel is fp32-matrix-pipe bound.

typedef float v2f __attribute__((ext_vector_type(2)));
typedef float v8f __attribute__((ext_vector_type(8)));

#define WMMA_F32_4(A, B, C)                                                    \
    __builtin_amdgcn_wmma_f32_16x16x4_f32(false, (A), false, (B), (short)0,    \
                                          (C), false, false)

namespace {
constexpr int kB = 4;
constexpr int kM = 8192;
constexpr int kN = 4096;
constexpr int kD = 3;
constexpr int kK = 16;
constexpr int kWavesPerBlock = 8;
constexpr int kTilesPerBatch = kM / 16;  // 512
constexpr float kEps = 1e-10f;
constexpr float kBig = 3.4e38f;  // finite upper clamp; med3(x,eps,BIG)==max(x,eps)
}  // namespace

__global__ __launch_bounds__(256) void polyharm_wmma_kernel(
    const float* __restrict__ x,   // [B, m, 3]
    const float* __restrict__ c,   // [B, n, 3]
    const float* __restrict__ w,   // [B, n, 16]
    const float* __restrict__ v,   // [B, 4, 16]
    float* __restrict__ out) {     // [B, m, 16]
    const int lane = threadIdx.x & 31;
    const int wave = threadIdx.x >> 5;
    const int tile = blockIdx.x * kWavesPerBlock + wave;
    const int b    = tile / kTilesPerBatch;
    const int m0   = (tile % kTilesPerBatch) * 16;

    const int r  = lane & 15;   // row (A/C) or column (B/C) index within tile
    const int hi = lane >> 4;   // which half-wave: selects K (A/B) / M+8 (C)

    const float* xb = x + (size_t)b * kM * kD;
    const float* cb = c + (size_t)b * kN * kD;
    const float* wb = w + (size_t)b * kN * kK;
    const float* vb = v + (size_t)b * (kD + 1) * kK;

    // ---- A-matrix: x_pad rows (x0, x1, x2, 1), 16x4 A-layout ----
    // VGPR0: lanes 0-15 -> K=0, lanes 16-31 -> K=2 ; VGPR1: K=1 / K=3.
    const float* xr = xb + (size_t)(m0 + r) * kD;
    const float x0 = xr[0], x1 = xr[1], x2 = xr[2];
    v2f ax;
    ax.x = hi ? x2 : x0;
    ax.y = hi ? 1.0f : x1;

    // ||x_row||^2 staged directly in C-layout: VGPR j, this lane -> M = j+8*hi.
    // Used as the C-input of the d2 WMMA (constant across N / lanes).
    const float xsq_own = x0 * x0 + x1 * x1 + x2 * x2;  // row m0 + (lane%16)
    v8f xsqv;
#pragma unroll
    for (int j = 0; j < 8; ++j) xsqv[j] = __shfl(xsq_own, j + 8 * hi, 32);

    // ---- Linear term: acc = x_pad @ v  (v[b] is exactly 4x16 = one B tile) ----
    v2f bv;
    bv.x = vb[(hi ? 2 : 0) * kK + r];
    bv.y = vb[(hi ? 3 : 1) * kK + r];
    v8f acc = {};
    acc = WMMA_F32_4(ax, bv, acc);

    // Wave-private LDS staging tile: C-layout phi -> A-layout reads.
    // No cross-wave sharing, so no barriers; per-wave LDS ops are in-order.
    __shared__ float phi_lds[kWavesPerBlock][16 * 16];
    float* pt = phi_lds[wave];

    for (int nb = 0; nb < kN; nb += 16) {
        // ---- B-matrix for d2: (-2c0, -2c1, -2c2, ||c||^2) columns ----
        const float* cr = cb + (size_t)(nb + r) * kD;
        const float c0 = cr[0], c1 = cr[1], c2 = cr[2];
        const float csq = c0 * c0 + c1 * c1 + c2 * c2;
        v2f bc;
        bc.x = hi ? (-2.0f * c2) : (-2.0f * c0);
        bc.y = hi ? csq : (-2.0f * c1);

        // One WMMA yields the full squared distance: xsq + (-2 x.c + csq).
        const v8f d2 = WMMA_F32_4(ax, bc, xsqv);

        // ---- phi = clamp(d2)^{3/2}: v_med3 + v_sqrt + v_mul per element ----
#pragma unroll
        for (int j = 0; j < 8; ++j) {
            const float t = __builtin_amdgcn_fmed3f(d2[j], kEps, kBig);
            pt[(j + 8 * hi) * 16 + r] = t * __builtin_amdgcn_sqrtf(t);
        }

        // ---- acc += phi_tile(16x16) @ w_tile(16x16) as 4 K=4 WMMAs ----
#pragma unroll
        for (int kb = 0; kb < 16; kb += 4) {
            // A-layout read: lane -> (M=r, K=kb+2*hi), VGPR1 = K+1 -> one b64 load.
            const v2f ap = *(const v2f*)&pt[r * 16 + kb + 2 * hi];
            const float* wrow = wb + (size_t)(nb + kb) * kK;
            v2f bw;
            bw.x = wrow[(hi ? 2 : 0) * kK + r];
            bw.y = wrow[(hi ? 3 : 1) * kK + r];
            acc = WMMA_F32_4(ap, bw, acc);
        }
    }

    // ---- Store C-layout accumulator: VGPR j, lane -> out[m0 + j + 8*hi][r] ----
    float* ob = out + ((size_t)b * kM + m0) * kK;
#pragma unroll
    for (int j = 0; j < 8; ++j) ob[(size_t)(j + 8 * hi) * kK + r] = acc[j];
}

extern "C" void kernel_launch(void* const* d_in, const int* in_sizes, int n_in,
                              void* d_out, int out_size, void* d_ws,
                              size_t ws_size, hipStream_t stream) {
    (void)in_sizes; (void)n_in; (void)d_ws; (void)ws_size; (void)out_size;
    const float* x = (const float*)d_in[0];
    const float* c = (const float*)d_in[1];
    const float* w = (const float*)d_in[2];
    const float* v = (const float*)d_in[3];
    float* out = (float*)d_out;

    const int total_waves = kB * kTilesPerBatch;            // 2048
    const int blocks = total_waves / kWavesPerBlock;        // 256
    polyharm_wmma_kernel<<<blocks, kWavesPerBlock * 32, 0, stream>>>(x, c, w, v,
                                                                     out);
}